// JAXModel_22179211116768
// MI455X (gfx1250) — compile-verified
//
#include <hip/hip_runtime.h>
#include <math.h>

// Problem constants (match reference)
#define B_   64
#define NX_  512
#define NY_  64
#define T_   256
#define NP_  (2*NX_ + NY_)

typedef float v2f __attribute__((ext_vector_type(2)));
typedef float v8f __attribute__((ext_vector_type(8)));
typedef int   v4i_ __attribute__((vector_size(16)));

// fp32 WMMA: D(16x16,f32) = A(16x4,f32) * B(4x16,f32) + C
__device__ __forceinline__ v8f wmma_f32_k4(v2f a, v2f b, v8f c) {
  return __builtin_amdgcn_wmma_f32_16x16x4_f32(false, a, false, b, (short)0, c,
                                               false, false);
}

__device__ __forceinline__ float softplus_f(float x) {
  return (x > 20.f) ? x : log1pf(expf(x));
}

// CDNA5 async global->LDS copies (ASYNCcnt), guarded so the sync path remains
// as fallback if the toolchain lacks the builtins.
#if defined(__gfx1250__) && \
    __has_builtin(__builtin_amdgcn_global_load_async_to_lds_b128) && \
    __has_builtin(__builtin_amdgcn_global_load_async_to_lds_b32)
#define USE_ASYNC_COPY 1
#else
#define USE_ASYNC_COPY 0
#endif

// Exact pointer types per hipcc diagnostics: AS1/AS3, non-const, typed pointee
// (hipcc prints address_space(1) as `__device__` in HIP mode).
#define GLOB_I32(p) ((__attribute__((address_space(1))) int*)(p))
#define GLOB_V4I(p) ((__attribute__((address_space(1))) v4i_*)(p))
#define LDS_I32(p)  ((__attribute__((address_space(3))) int*)(p))
#define LDS_V4I(p)  ((__attribute__((address_space(3))) v4i_*)(p))

#if USE_ASYNC_COPY
__device__ __forceinline__ void async_wait0() {
#if __has_builtin(__builtin_amdgcn_s_wait_asynccnt)
  __builtin_amdgcn_s_wait_asynccnt(0);
#else
  asm volatile("s_wait_asynccnt 0x0" ::: "memory");
#endif
}
#endif

// ---------------------------------------------------------------------------
// 0) zero the llh accumulator
__global__ void zero_kernel(float* out) { out[0] = 0.f; }

// ---------------------------------------------------------------------------
// 1) unscale params, softplus, build x0, Dinv = 1/(1+h*d), sig
__global__ void params_kernel(const float* __restrict__ ts,
                              const float* __restrict__ p,
                              const int*   __restrict__ pscale,
                              const float* __restrict__ k,
                              float* __restrict__ x0,
                              float* __restrict__ Dinv,
                              float* __restrict__ sig) {
  int idx = blockIdx.x * blockDim.x + threadIdx.x;
  if (idx >= B_ * NP_) return;
  const float h = (ts[1] - ts[0]) * 0.25f;   // DT / N_SUB
  int b = idx / NP_, j = idx % NP_;
  float v = p[idx];
  int s = pscale[idx];
  float ps = (s == 0) ? v : ((s == 1) ? expf(v) : exp10f(v));
  float sp = softplus_f(ps);
  if (j < NX_) {
    x0[b * NX_ + j] = sp + k[j];
  } else if (j < 2 * NX_) {
    float d = sp + 0.1f;
    Dinv[b * NX_ + (j - NX_)] = 1.0f / (1.0f + h * d);
  } else {
    sig[b * NY_ + (j - 2 * NX_)] = sp + 1e-3f;
  }
}

// ---------------------------------------------------------------------------
// GEMM tiling: 64x128 block, BK=32, 256 threads (8 waves).
// Wave w owns a 16x64 slab: rows 16*(w&3), cols 64*(w>>2); 4 accumulators.
// B is stored TRANSPOSED in LDS so each operand is one contiguous b64 load.
#define BM 64
#define BN 128
#define BK 32
#define LDP (BK + 4)   // 144B row stride: 16B-aligned rows, even (v2f), 36r%64 conflict-free

// 2) Build M = (I + eps*G + eps^2*G*G) * diag(Dinv), G = diag(Dinv)*A_off.
//    Sync staging here: operands are scaled by Dinv while being staged.
__global__ __launch_bounds__(256)
void build_M_kernel(const float* __restrict__ ts,
                    const float* __restrict__ Aoff,
                    const float* __restrict__ Dinv,
                    float* __restrict__ M) {
  __shared__ float Als[BM][LDP];      // A[m][k]   (= G row block)
  __shared__ float Bls[BN][LDP];      // B^T: [n][k] (= G col block, transposed)
  const int bz = blockIdx.z;
  const float* Db = Dinv + bz * NX_;
  float* Mb = M + (size_t)bz * NX_ * NX_;
  const float h   = (ts[1] - ts[0]) * 0.25f;
  const float eps = 0.01f * h;
  const int m0 = blockIdx.y * BM, n0 = blockIdx.x * BN;
  const int tid = threadIdx.x, lane = tid & 31, wv = tid >> 5;
  const int hl = lane >> 4, l15 = lane & 15;
  const int mt = (wv & 3) * 16, nt = (wv >> 2) * 64;
  v8f acc[4];
  #pragma unroll
  for (int q = 0; q < 4; ++q) acc[q] = (v8f){};
  for (int kb = 0; kb < NX_; kb += BK) {
    #pragma unroll
    for (int i = 0; i < 8; ++i) {           // A: G[m0..+64, kb..+32]
      int e = tid + i * 256; int r = e >> 5, c = e & 31;
      Als[r][c] = Db[m0 + r] * Aoff[(size_t)(m0 + r) * NX_ + kb + c];
    }
    #pragma unroll
    for (int i = 0; i < 16; ++i) {          // B^T: G[kb..+32, n0..+128]
      int e = tid + i * 256; int r = e >> 7, c = e & 127;
      Bls[c][r] = Db[kb + r] * Aoff[(size_t)(kb + r) * NX_ + n0 + c];
    }
    __syncthreads();
    v2f areg[8], breg[8][4];
    #pragma unroll
    for (int s = 0; s < 8; ++s) {           // operand preload: ds_load_b64
      int k0 = s * 4 + hl * 2;
      areg[s] = *(const v2f*)&Als[mt + l15][k0];
      #pragma unroll
      for (int q = 0; q < 4; ++q)
        breg[s][q] = *(const v2f*)&Bls[nt + q * 16 + l15][k0];
    }
    #pragma unroll
    for (int s = 0; s < 8; ++s) {           // 32 WMMAs
      #pragma unroll
      for (int q = 0; q < 4; ++q)
        acc[q] = wmma_f32_k4(areg[s], breg[s][q], acc[q]);
    }
    __syncthreads();
  }
  #pragma unroll
  for (int v = 0; v < 8; ++v) {
    int row = m0 + mt + v + hl * 8;
    float dr = Db[row];
    #pragma unroll
    for (int q = 0; q < 4; ++q) {
      int col = n0 + nt + q * 16 + l15;
      float g = dr * Aoff[(size_t)row * NX_ + col];
      float mv = (((row == col) ? 1.f : 0.f) + eps * g + eps * eps * acc[q][v])
                 * Db[col];
      Mb[(size_t)row * NX_ + col] = mv;
    }
  }
}

// 3) batched matrix square: Y_b = X_b * X_b  (used for M2 and M4).
//    Async global->LDS staging with LDS double-buffering: prefetch K-block
//    kb+BK into the alternate buffer while WMMAs run on the current one.
#if USE_ASYNC_COPY
#define STAGE_ASYNC(kb_, bf_)                                                  \
  do {                                                                         \
    _Pragma("unroll")                                                          \
    for (int i = 0; i < 2; ++i) {         /* A tile: b128 per lane */          \
      int gidx = tid + i * 256; int r = gidx >> 3, c4 = (gidx & 7) * 4;        \
      __builtin_amdgcn_global_load_async_to_lds_b128(                          \
          GLOB_V4I(Xb + (size_t)(m0 + r) * NX_ + (kb_) + c4),                  \
          LDS_V4I(&Als[bf_][r][c4]), 0, 0);                                    \
    }                                                                          \
    _Pragma("unroll")                                                          \
    for (int i = 0; i < 16; ++i) {        /* B tile: transpose via b32 */      \
      int e = tid + i * 256; int r = e >> 7, c = e & 127;                      \
      __builtin_amdgcn_global_load_async_to_lds_b32(                           \
          GLOB_I32(Xb + (size_t)((kb_) + r) * NX_ + n0 + c),                   \
          LDS_I32(&Bls[bf_][c][r]), 0, 0);                                     \
    }                                                                          \
  } while (0)
#endif

__global__ __launch_bounds__(256)
void gemm_square_kernel(const float* __restrict__ X, float* __restrict__ Y) {
#if USE_ASYNC_COPY
  __shared__ float Als[2][BM][LDP];
  __shared__ float Bls[2][BN][LDP];
#else
  __shared__ float Als[1][BM][LDP];
  __shared__ float Bls[1][BN][LDP];
#endif
  const int bz = blockIdx.z;
  const float* Xb = X + (size_t)bz * NX_ * NX_;
  float* Yb = Y + (size_t)bz * NX_ * NX_;
  const int m0 = blockIdx.y * BM, n0 = blockIdx.x * BN;
  const int tid = threadIdx.x, lane = tid & 31, wv = tid >> 5;
  const int hl = lane >> 4, l15 = lane & 15;
  const int mt = (wv & 3) * 16, nt = (wv >> 2) * 64;
  v8f acc[4];
  #pragma unroll
  for (int q = 0; q < 4; ++q) acc[q] = (v8f){};

#if USE_ASYNC_COPY
  int cur = 0;
  STAGE_ASYNC(0, 0);
  async_wait0();
  __syncthreads();
  for (int kb = 0; kb < NX_; kb += BK) {
    if (kb + BK < NX_) STAGE_ASYNC(kb + BK, cur ^ 1);  // prefetch next block
    v2f areg[8], breg[8][4];
    #pragma unroll
    for (int s = 0; s < 8; ++s) {
      int k0 = s * 4 + hl * 2;
      areg[s] = *(const v2f*)&Als[cur][mt + l15][k0];
      #pragma unroll
      for (int q = 0; q < 4; ++q)
        breg[s][q] = *(const v2f*)&Bls[cur][nt + q * 16 + l15][k0];
    }
    #pragma unroll
    for (int s = 0; s < 8; ++s) {
      #pragma unroll
      for (int q = 0; q < 4; ++q)
        acc[q] = wmma_f32_k4(areg[s], breg[s][q], acc[q]);
    }
    async_wait0();        // next block's LDS data landed (this wave's part)
    __syncthreads();      // ...and everyone else's; current buffer free
    cur ^= 1;
  }
#else
  for (int kb = 0; kb < NX_; kb += BK) {
    #pragma unroll
    for (int i = 0; i < 8; ++i) {
      int e = tid + i * 256; int r = e >> 5, c = e & 31;
      Als[0][r][c] = Xb[(size_t)(m0 + r) * NX_ + kb + c];
    }
    #pragma unroll
    for (int i = 0; i < 16; ++i) {
      int e = tid + i * 256; int r = e >> 7, c = e & 127;
      Bls[0][c][r] = Xb[(size_t)(kb + r) * NX_ + n0 + c];
    }
    __syncthreads();
    v2f areg[8], breg[8][4];
    #pragma unroll
    for (int s = 0; s < 8; ++s) {
      int k0 = s * 4 + hl * 2;
      areg[s] = *(const v2f*)&Als[0][mt + l15][k0];
      #pragma unroll
      for (int q = 0; q < 4; ++q)
        breg[s][q] = *(const v2f*)&Bls[0][nt + q * 16 + l15][k0];
    }
    #pragma unroll
    for (int s = 0; s < 8; ++s) {
      #pragma unroll
      for (int q = 0; q < 4; ++q)
        acc[q] = wmma_f32_k4(areg[s], breg[s][q], acc[q]);
    }
    __syncthreads();
  }
#endif
  #pragma unroll
  for (int v = 0; v < 8; ++v) {
    int row = m0 + mt + v + hl * 8;
    #pragma unroll
    for (int q = 0; q < 4; ++q) {
      int col = n0 + nt + q * 16 + l15;
      Yb[(size_t)row * NX_ + col] = acc[q][v];
    }
  }
}

// ---------------------------------------------------------------------------
// 4) Fused time scan + observables + NLL. One workgroup (32 waves) per
//    condition; x double-buffered in LDS; M4_b (1MB) streamed from L2
//    (all 64MB of M4 is L2-resident on a 192MB L2). float4 loads throughout.
__global__ __launch_bounds__(1024)
void scan_nll_kernel(const float* __restrict__ M4,
                     const float* __restrict__ x0,
                     const float* __restrict__ sig,
                     const float* __restrict__ C,
                     const float* __restrict__ my,
                     float* __restrict__ llh) {
  __shared__ float xbuf[2][NX_];
  __shared__ float sigl[NY_], logl[NY_];
  const int b = blockIdx.x, tid = threadIdx.x, lane = tid & 31, wv = tid >> 5;
  if (tid < NX_) xbuf[0][tid] = x0[b * NX_ + tid];
  if (tid < NY_) {
    float s = sig[b * NY_ + tid];
    sigl[tid] = s;
    logl[tid] = logf(6.28318530717958647692f * s * s);
  }
  __syncthreads();
  const float* M4b = M4 + (size_t)b * NX_ * NX_;
  float acc = 0.f;
  for (int t = 0; t < T_; ++t) {
    const float4* xc4 = (const float4*)xbuf[t & 1];
    // obs + Gaussian NLL for this timepoint (llh = sum 0.5*(log(2*pi*s^2)+r^2))
    for (int y = wv; y < NY_; y += 32) {
      const float4* Cy4 = (const float4*)(C + (size_t)y * NX_);
      float s = 0.f;
      #pragma unroll
      for (int i = 0; i < 4; ++i) {
        float4 cv = Cy4[lane + 32 * i];
        float4 xv = xc4[lane + 32 * i];
        s += cv.x * xv.x + cv.y * xv.y + cv.z * xv.z + cv.w * xv.w;
      }
      #pragma unroll
      for (int o = 16; o > 0; o >>= 1) s += __shfl_xor(s, o);
      if (lane == 0) {
        float r = (s - my[((size_t)b * T_ + t) * NY_ + y]) / sigl[y];
        acc += 0.5f * (logl[y] + r * r);
      }
    }
    if (t + 1 < T_) {                        // x_{t+1} = M4 * x_t
      float* xn = xbuf[(t + 1) & 1];
      for (int r = wv; r < NX_; r += 32) {
        const float4* Mr4 = (const float4*)(M4b + (size_t)r * NX_);
        float s = 0.f;
        #pragma unroll
        for (int i = 0; i < 4; ++i) {
          float4 mv = Mr4[lane + 32 * i];
          float4 xv = xc4[lane + 32 * i];
          s += mv.x * xv.x + mv.y * xv.y + mv.z * xv.z + mv.w * xv.w;
        }
        #pragma unroll
        for (int o = 16; o > 0; o >>= 1) s += __shfl_xor(s, o);
        if (lane == 0) xn[r] = s;
      }
    }
    __syncthreads();
  }
  if (lane == 0) atomicAdd(llh, acc);
}

// ---------------------------------------------------------------------------
extern "C" void kernel_launch(void* const* d_in, const int* in_sizes, int n_in,
                              void* d_out, int out_size, void* d_ws, size_t ws_size,
                              hipStream_t stream) {
  const float* ts     = (const float*)d_in[0];
  const float* p      = (const float*)d_in[1];
  const float* k      = (const float*)d_in[2];
  const float* my     = (const float*)d_in[3];
  const float* Aoff   = (const float*)d_in[4];
  const float* C      = (const float*)d_in[5];
  const int*   pscale = (const int*)d_in[6];
  float* out = (float*)d_out;

  // Workspace layout (floats): x0 | Dinv | sig | bufA(64MB) | bufB(64MB)
  float* ws   = (float*)d_ws;
  float* x0   = ws;
  float* Dinv = x0 + (size_t)B_ * NX_;
  float* sig  = Dinv + (size_t)B_ * NX_;
  float* bufA = sig + (size_t)B_ * NY_;                 // M, then M4
  float* bufB = bufA + (size_t)B_ * NX_ * NX_;          // M2

  zero_kernel<<<1, 1, 0, stream>>>(out);
  params_kernel<<<(B_ * NP_ + 255) / 256, 256, 0, stream>>>(ts, p, pscale, k,
                                                            x0, Dinv, sig);
  dim3 g(NX_ / BN, NX_ / BM, B_);
  build_M_kernel<<<g, 256, 0, stream>>>(ts, Aoff, Dinv, bufA);   // M
  gemm_square_kernel<<<g, 256, 0, stream>>>(bufA, bufB);         // M2 = M*M
  gemm_square_kernel<<<g, 256, 0, stream>>>(bufB, bufA);         // M4 = M2*M2
  scan_nll_kernel<<<B_, 1024, 0, stream>>>(bufA, x0, sig, C, my, out);
}